// DynamicUpsamplingFilter_3C_4286377361926
// MI455X (gfx1250) — compile-verified
//
#include <hip/hip_runtime.h>

typedef __attribute__((ext_vector_type(2))) float v2f;
typedef __attribute__((ext_vector_type(8))) float v8f;

#define Bc     4
#define Cc     3
#define Hc     128
#define Wc     128
#define Rr     16
#define NFt    81
#define HWc    (Hc * Wc)
#define TILE_W 8
#define FR     (NFt * Rr)     // 1296 (f,r) rows
#define FRP    1300           // padded stride (1300 % 64 == 20) for bank spread
#define XT_W   (TILE_W + 8)   // 16: 8 pixels + 2*4 halo
#define KFULL  20             // chunks with all four f-taps valid (f = 0..79)

// One block = 256 threads = 8 wave32 = 8 consecutive w-pixels at one (b,h).
// Per wave: D[16x16] (M=r, N=c; c<3 used) = A[r,f] (filters^T) x B[f,c]
// (9x9 patches of x), accumulated over 21 V_WMMA_F32_16X16X4_F32 (K=81 -> 20
// full chunks + peeled tail). Inner loop is fully branchless (cndmask only),
// keeping EXEC all-ones around the WMMAs.
__global__ __launch_bounds__(256) void duf_wmma_kernel(
    const float* __restrict__ xin,     // [B,3,H,W]
    const float* __restrict__ filt,    // [B,81,16,H,W]
    float* __restrict__ out) {         // [B,48,H,W]
  __shared__ float lds_f[TILE_W * FRP];    // [w'][f*16 + r]
  __shared__ float lds_x[Cc * 9 * XT_W];   // [c][dy][dx'] zero-padded halo
  __shared__ float lds_d[Cc * Rr * TILE_W];// [c*16+r][w'] staged output

  const int tid = threadIdx.x;
  const int blk = blockIdx.x;          // b*(H*16) + h*16 + wt
  const int b   = blk >> 11;           // / (128*16)
  const int rem = blk & 2047;
  const int h   = rem >> 4;
  const int w0  = (rem & 15) * TILE_W;

  // ---- cooperative load: x halo tile (zeros outside the image) ----
  for (int i = tid; i < Cc * 9 * XT_W; i += 256) {
    const int c  = i / (9 * XT_W);
    const int r2 = i - c * (9 * XT_W);
    const int dy = r2 >> 4;            // XT_W == 16
    const int dx = r2 & 15;
    const int gy = h - 4 + dy;
    const int gx = w0 - 4 + dx;
    float v = 0.0f;
    if ((unsigned)gy < (unsigned)Hc && (unsigned)gx < (unsigned)Wc)
      v = xin[(((size_t)b * Cc + c) * Hc + gy) * Wc + gx];
    lds_x[i] = v;
  }

  // ---- cooperative load: filters, b128 loads coalesced along w, transposed ----
  // i -> (fr, q): lanes pair up on 32B-contiguous runs; 16 (f,r) rows per
  // wave-load. Transposed scatter into LDS is bank-conflict-free (stride 1300).
  const size_t fbase = (size_t)b * FR * HWc + (size_t)h * Wc + (size_t)w0;
  const float4* __restrict__ filt4 = (const float4*)filt;
  for (int i = tid; i < FR * 2; i += 256) {
    const int fr = i >> 1;
    const int q  = i & 1;
    const float4 v4 = filt4[(fbase + (size_t)fr * HWc) / 4 + q];
    const int w4 = q * 4;
    lds_f[(w4 + 0) * FRP + fr] = v4.x;
    lds_f[(w4 + 1) * FRP + fr] = v4.y;
    lds_f[(w4 + 2) * FRP + fr] = v4.z;
    lds_f[(w4 + 3) * FRP + fr] = v4.w;
  }
  __syncthreads();

  // ---- per-wave WMMA accumulation over f (branchless) ----
  const int lane = tid & 31;
  const int p    = tid >> 5;           // wave id == pixel within tile
  const int lo16 = lane & 15;          // M index (r) for A; N index (c) for B
  const int hi   = lane >> 4;          // K half select (K+2, K+3 slots)
  const bool bval = (lo16 < Cc);
  const int  cc   = bval ? lo16 : (Cc - 1);       // clamped channel
  const float* fp = &lds_f[p * FRP];
  const int  aoff = hi * 32 + lo16;    // A dword offset within 64-dword chunk
  const int  cb   = cc * (9 * XT_W) + p;          // patch base for this lane

  v8f acc = {};
#pragma unroll
  for (int k = 0; k < KFULL; ++k) {
    // A-matrix: lane lo16 = r; K slots (VGPR0 -> K0/K2, VGPR1 -> K1/K3).
    v2f a;
    a.x = fp[64 * k + aoff];
    a.y = fp[64 * k + 16 + aoff];

    // B-matrix: compile-time (kh,kw) offsets, hi selects the K+2 pair.
    const int fa0 = 4 * k, fa1 = fa0 + 1, fa2 = fa0 + 2, fa3 = fa0 + 3;
    const int o0 = hi ? ((fa2 / 9) * XT_W + fa2 % 9)
                      : ((fa0 / 9) * XT_W + fa0 % 9);
    const int o1 = hi ? ((fa3 / 9) * XT_W + fa3 % 9)
                      : ((fa1 / 9) * XT_W + fa1 % 9);
    const float b0 = lds_x[cb + o0];
    const float b1 = lds_x[cb + o1];
    v2f bb;
    bb.x = bval ? b0 : 0.0f;
    bb.y = bval ? b1 : 0.0f;

    // 8 args: (neg_a, A, neg_b, B, c_mod, C, reuse_a, reuse_b)
    acc = __builtin_amdgcn_wmma_f32_16x16x4_f32(false, a, false, bb,
                                                (short)0, acc, false, false);
  }

  // ---- peeled tail chunk: only f = 80 (slot K0, hi == 0) is live ----
  {
    const float at = fp[64 * KFULL + lo16];          // f = 80 row
    const float bt = lds_x[cb + (8 * XT_W + 8)];     // kh = 8, kw = 8
    v2f a, bb;
    a.x  = hi ? 0.0f : at;
    a.y  = 0.0f;
    bb.x = (bval && !hi) ? bt : 0.0f;
    bb.y = 0.0f;
    acc = __builtin_amdgcn_wmma_f32_16x16x4_f32(false, a, false, bb,
                                                (short)0, acc, false, false);
  }

  // ---- stage D to LDS, then cooperative coalesced store ----
  if (bval) {
    const int c = lo16;
#pragma unroll
    for (int v = 0; v < 8; ++v) {
      const int r = v + (hi ? 8 : 0);  // D: vgpr v holds rows M=v / M=v+8
      lds_d[(c * Rr + r) * TILE_W + p] = acc[v];
    }
  }
  __syncthreads();

  const size_t obase = (size_t)b * (Cc * Rr) * HWc + (size_t)h * Wc + w0;
  for (int i = tid; i < Cc * Rr * TILE_W; i += 256) {
    const int cr = i >> 3;             // TILE_W == 8
    const int wp = i & 7;
    out[obase + (size_t)cr * HWc + wp] = lds_d[i];
  }
}

extern "C" void kernel_launch(void* const* d_in, const int* in_sizes, int n_in,
                              void* d_out, int out_size, void* d_ws, size_t ws_size,
                              hipStream_t stream) {
  (void)in_sizes; (void)n_in; (void)out_size; (void)d_ws; (void)ws_size;
  const float* x    = (const float*)d_in[0];  // [4,3,128,128]
  const float* filt = (const float*)d_in[1];  // [4,81,16,128,128]
  float* out        = (float*)d_out;          // [4,48,128,128]

  const int blocks = Bc * Hc * (Wc / TILE_W); // 8192
  duf_wmma_kernel<<<blocks, 256, 0, stream>>>(x, filt, out);
}